// TokenSelection_20426864460130
// MI455X (gfx1250) — compile-verified
//
#include <hip/hip_runtime.h>

typedef __attribute__((ext_vector_type(2))) float v2f;
typedef __attribute__((ext_vector_type(4))) float v4f;
typedef __attribute__((ext_vector_type(8))) float v8f;

#define NB   8          // batches
#define TN   18432      // 32*576 tokens per batch
#define EE   1024       // embed dim
#define TOPK 2048
#define NTOK (NB * TN)  // 147456 total tokens

// ---------------------------------------------------------------------------
// Kernel A: scores[token] = dot(x[token], w) via V_WMMA_F32_16X16X4_F32.
// Per wave: 16 tokens live in B's N dimension; A row M=0 carries the w chunk
// (lanes 0 and 16 per the 16x4 f32 A layout), other A rows are zero.
// D[0][n] = score of token n -> c[0] on lanes 0..15, coalesced store.
// Two accumulators break the WMMA->WMMA accumulation dependency.
// ---------------------------------------------------------------------------
__global__ __launch_bounds__(256) void score_kernel(const float* __restrict__ x,
                                                    const float* __restrict__ w,
                                                    float* __restrict__ scores) {
  const int lane      = threadIdx.x & 31;
  const int wave      = threadIdx.x >> 5;
  const int tokenBase = blockIdx.x * 128 + wave * 16;   // 8 waves * 16 tokens
  const int khalf     = (lane >> 4) * 2;                // lanes 0-15: K+0/1, 16-31: K+2/3
  const bool isRow0   = (lane & 15) == 0;               // A-matrix row M=0 holders

  const float* xr = x + (size_t)(tokenBase + (lane & 15)) * EE + khalf;
  const float* wr = w + khalf;

  v8f c0 = {};
  v8f c1 = {};
  for (int k = 0; k < EE; k += 8) {
    v2f b0  = *(const v2f*)(xr + k);        // B[k..k+1][n=lane&15]
    v2f b1  = *(const v2f*)(xr + k + 4);
    v2f wv0 = *(const v2f*)(wr + k);        // broadcast load, L0 hit
    v2f wv1 = *(const v2f*)(wr + k + 4);
    v2f a0 = { isRow0 ? wv0.x : 0.0f, isRow0 ? wv0.y : 0.0f };
    v2f a1 = { isRow0 ? wv1.x : 0.0f, isRow0 ? wv1.y : 0.0f };
    c0 = __builtin_amdgcn_wmma_f32_16x16x4_f32(false, a0, false, b0,
                                               (short)0, c0, false, false);
    c1 = __builtin_amdgcn_wmma_f32_16x16x4_f32(false, a1, false, b1,
                                               (short)0, c1, false, false);
  }
  // D[0][n]: VGPR 0, lanes 0..15 hold scores of tokens tokenBase+0..15.
  if (lane < 16) scores[tokenBase + lane] = c0[0] + c1[0];
}

// Map float bits to monotonically-increasing unsigned order.
__device__ __forceinline__ unsigned f2ord(float f) {
  unsigned u = __float_as_uint(f);
  return (u & 0x80000000u) ? ~u : (u | 0x80000000u);
}

// ---------------------------------------------------------------------------
// Kernel B: exact top-2048 per batch, fully deterministic.
// One 1024-thread workgroup per batch; 18 contiguous elements per thread.
//  1) 4x 8-bit MSB-first radix histogram -> exact threshold + tie quota
//  2) packed (greater|equal) Hillis-Steele scan -> deterministic slots,
//     ties resolved by ascending index (matches jax.lax.top_k)
//  3) bitonic sort of 2048 keys (~score_bits:idx) -> descending-score order
// ---------------------------------------------------------------------------
__global__ __launch_bounds__(1024) void topk_kernel(const float* __restrict__ scores,
                                                    int* __restrict__ sel) {
  const int b   = blockIdx.x;
  const int tid = threadIdx.x;
  const float* s = scores + (size_t)b * TN;

  __shared__ unsigned hist[256];
  __shared__ unsigned scanBuf[1024];
  __shared__ unsigned long long keyArr[TOPK];
  __shared__ unsigned sh_prefix, sh_need;

  // Load + order-map 18 contiguous elements per thread.
  unsigned uv[18];
  const int base = tid * 18;
#pragma unroll
  for (int j = 0; j < 18; ++j) uv[j] = f2ord(s[base + j]);

  // ---- radix threshold search ----
  unsigned prefix = 0, prefMask = 0, need = TOPK;
  for (int pass = 0; pass < 4; ++pass) {
    const int shift = 24 - 8 * pass;
    if (tid < 256) hist[tid] = 0;
    __syncthreads();
#pragma unroll
    for (int j = 0; j < 18; ++j) {
      unsigned u = uv[j];
      if ((u & prefMask) == prefix)
        atomicAdd(&hist[(u >> shift) & 0xFF], 1u);
    }
    __syncthreads();
    if (tid == 0) {
      unsigned cum = 0; int bin = 0;
      for (int i = 255; i >= 0; --i) {
        if (cum + hist[i] >= need) { bin = i; break; }
        cum += hist[i];
      }
      sh_prefix = prefix | ((unsigned)bin << shift);
      sh_need   = need - cum;   // remaining picks inside chosen bin
    }
    __syncthreads();
    prefix = sh_prefix;
    need   = sh_need;
    prefMask |= 0xFFu << shift;
    __syncthreads();
  }
  const unsigned ut = prefix;   // exact threshold (order-mapped)

  // ---- init key slots ----
  keyArr[tid]        = ~0ull;
  keyArr[tid + 1024] = ~0ull;

  // ---- packed counts + scan ----
  unsigned packed = 0;
#pragma unroll
  for (int j = 0; j < 18; ++j) {
    if (uv[j] > ut)       packed += 0x10000u;
    else if (uv[j] == ut) packed += 1u;
  }
  scanBuf[tid] = packed;
  __syncthreads();
  for (int off = 1; off < 1024; off <<= 1) {
    unsigned v   = scanBuf[tid];
    unsigned add = (tid >= off) ? scanBuf[tid - off] : 0u;
    __syncthreads();
    scanBuf[tid] = v + add;
    __syncthreads();
  }
  const unsigned incl  = scanBuf[tid];
  const unsigned total = scanBuf[1023];
  const unsigned Gtot  = total >> 16;          // strictly-greater count (< TOPK)
  const unsigned excl  = incl - packed;
  unsigned gpos = excl >> 16;
  unsigned epos = excl & 0xFFFFu;

#pragma unroll
  for (int j = 0; j < 18; ++j) {
    unsigned u = uv[j];
    if (u > ut) {
      keyArr[gpos] = ((unsigned long long)(~u) << 32) | (unsigned)(base + j);
      ++gpos;
    } else if (u == ut) {
      if (Gtot + epos < TOPK)   // lowest-index ties win deterministically
        keyArr[Gtot + epos] = ((unsigned long long)(~u) << 32) | (unsigned)(base + j);
      ++epos;
    }
  }
  __syncthreads();

  // ---- bitonic sort ascending on key => score desc, index asc ----
  for (unsigned k = 2; k <= TOPK; k <<= 1) {
    for (unsigned j = k >> 1; j > 0; j >>= 1) {
#pragma unroll
      for (int r = 0; r < 2; ++r) {
        unsigned ii = tid + (unsigned)r * 1024u;
        unsigned jj = ii ^ j;
        if (jj > ii) {
          unsigned long long a = keyArr[ii];
          unsigned long long c = keyArr[jj];
          if ((a > c) == ((ii & k) == 0)) { keyArr[ii] = c; keyArr[jj] = a; }
        }
      }
      __syncthreads();
    }
  }

  sel[b * TOPK + tid]        = (int)(keyArr[tid] & 0xFFFFFFFFull);
  sel[b * TOPK + tid + 1024] = (int)(keyArr[tid + 1024] & 0xFFFFFFFFull);
}

// ---------------------------------------------------------------------------
// Kernel C: gather selected rows, float4 (GLOBAL_LOAD/STORE_B128) copies.
// One block per output row: 256 lanes x 16B = 4 KB row.
// ---------------------------------------------------------------------------
__global__ __launch_bounds__(256) void gather_kernel(const float* __restrict__ x,
                                                     const int* __restrict__ sel,
                                                     float* __restrict__ out) {
  const int id  = blockIdx.x;        // 0 .. NB*TOPK-1
  const int b   = id >> 11;
  const int row = sel[id];           // local index within batch, 0..TN-1
  const v4f* src = (const v4f*)(x + ((size_t)b * TN + (size_t)row) * EE);
  v4f*       dst = (v4f*)(out + (size_t)id * EE);
  dst[threadIdx.x] = src[threadIdx.x];
}

extern "C" void kernel_launch(void* const* d_in, const int* in_sizes, int n_in,
                              void* d_out, int out_size, void* d_ws, size_t ws_size,
                              hipStream_t stream) {
  (void)in_sizes; (void)n_in; (void)out_size; (void)ws_size;
  const float* x = (const float*)d_in[0];
  const float* w = (const float*)d_in[1];
  // d_in[2] (bias) shifts every score equally -> cannot change top-k; ignored.
  float* out    = (float*)d_out;
  float* scores = (float*)d_ws;                                   // 147456 f32
  int*   sel    = (int*)((char*)d_ws + (size_t)NTOK * sizeof(float)); // 8*2048 i32

  score_kernel<<<NTOK / 128, 256, 0, stream>>>(x, w, scores);
  topk_kernel<<<NB, 1024, 0, stream>>>(scores, sel);
  gather_kernel<<<NB * TOPK, 256, 0, stream>>>(x, sel, out);
}